// SO3reparameterize_6889127542979
// MI455X (gfx1250) — compile-verified
//
#include <hip/hip_runtime.h>

#define TPB 256
#define NV4 (TPB * 9 / 4)  // 576 float4 per block tile

// ---- CDNA5 async global<->LDS path (gate on builtin availability) ----
#if defined(__AMDGCN__) &&                                          \
    __has_builtin(__builtin_amdgcn_global_load_async_to_lds_b128) && \
    __has_builtin(__builtin_amdgcn_global_store_async_from_lds_b128) && \
    __has_builtin(__builtin_amdgcn_s_wait_asynccnt)
#define USE_ASYNC 1
#else
#define USE_ASYNC 0
#endif

#if USE_ASYNC
#define GLOBAL_AS __attribute__((address_space(1)))
#define LDS_AS __attribute__((address_space(3)))
typedef __attribute__((__vector_size__(4 * sizeof(int)))) int i32x4;
typedef GLOBAL_AS i32x4* gvec_p;
typedef LDS_AS i32x4* lvec_p;

__device__ __forceinline__ void async_g2l_b128(const void* g, void* l) {
  // global -> LDS, 16B per lane, tracked by ASYNCcnt
  __builtin_amdgcn_global_load_async_to_lds_b128((gvec_p)g, (lvec_p)l, 0, 0);
}
__device__ __forceinline__ void async_l2g_b128(void* g, const void* l) {
  // LDS -> global, 16B per lane, tracked by ASYNCcnt
  __builtin_amdgcn_global_store_async_from_lds_b128((gvec_p)g, (lvec_p)l, 0,
                                                    0);
}
__device__ __forceinline__ void wait_async0() {
  __builtin_amdgcn_s_wait_asynccnt(0);
}
#endif

// Main kernel: one thread per batch element b, loops over the sample dim.
// mu tile and z tiles are staged through LDS so all global traffic is
// 128-bit, fully coalesced, and 16B-aligned. On CDNA5 the staging uses the
// async global<->LDS data movers (no VGPR round-trip, ASYNCcnt-tracked).
__global__ __launch_bounds__(TPB) void so3_reparam_tiled(
    const float* __restrict__ mu,   // [B, 3, 3]
    const float* __restrict__ v,    // [N, B, 3]
    float* __restrict__ out,        // [N, B, 3, 3]
    int B, int nsamp) {
  __shared__ __align__(16) float lds[TPB * 9];  // 9216 B, reused: mu in, z out
  const int tid = threadIdx.x;
  const int b0 = blockIdx.x * TPB;
  const int b = b0 + tid;

  // ---- stage mu tile: contiguous 9216 B, base (b0*36) is 16B-aligned ----
  {
    const float4* __restrict__ gmu4 =
        reinterpret_cast<const float4*>(mu + (size_t)b0 * 9);
    float4* lds4 = reinterpret_cast<float4*>(lds);
#pragma unroll
    for (int i = 0; i < 3; ++i) {
      int idx = tid + i * TPB;
      if (idx < NV4) {
#if USE_ASYNC
        async_g2l_b128(&gmu4[idx], &lds4[idx]);
#else
        lds4[idx] = gmu4[idx];
#endif
      }
    }
  }
#if USE_ASYNC
  wait_async0();
#endif
  __syncthreads();
  float m[9];
#pragma unroll
  for (int k = 0; k < 9; ++k) m[k] = lds[tid * 9 + k];

  for (int n = 0; n < nsamp; ++n) {
    // ---- load v[n, b, :] (3 consecutive floats -> b96) ----
    const size_t vb = ((size_t)n * B + b) * 3;
    float vx = v[vb + 0];
    float vy = v[vb + 1];
    float vz = v[vb + 2];
    if (n + 1 < nsamp) {
      // prefetch next sample's v for this lane (global_prefetch_b8)
      __builtin_prefetch(&v[vb + (size_t)B * 3], 0, 0);
    }

    // ---- Rodrigues: R = c*I + s*K + (1-c) u u^T ----
    float d = fmaf(vx, vx, fmaf(vy, vy, vz * vz));
    float inv = __frsqrt_rn(d);   // 1/theta
    float theta = d * inv;        // |v|
    float ux = vx * inv, uy = vy * inv, uz = vz * inv;
    float s = __sinf(theta);
    float c = __cosf(theta);
    float oc = 1.0f - c;

    float ocx = oc * ux, ocy = oc * uy, ocz = oc * uz;
    float R00 = fmaf(ocx, ux, c);
    float R01 = fmaf(ocx, uy, -s * uz);
    float R02 = fmaf(ocx, uz, s * uy);
    float R10 = fmaf(ocy, ux, s * uz);
    float R11 = fmaf(ocy, uy, c);
    float R12 = fmaf(ocy, uz, -s * ux);
    float R20 = fmaf(ocz, ux, -s * uy);
    float R21 = fmaf(ocz, uy, s * ux);
    float R22 = fmaf(ocz, uz, c);

    // ---- z = mu @ R (3x3 * 3x3, 27 FMAs) ----
    float z[9];
#pragma unroll
    for (int i = 0; i < 3; ++i) {
      float a0 = m[i * 3 + 0], a1 = m[i * 3 + 1], a2 = m[i * 3 + 2];
      z[i * 3 + 0] = fmaf(a0, R00, fmaf(a1, R10, a2 * R20));
      z[i * 3 + 1] = fmaf(a0, R01, fmaf(a1, R11, a2 * R21));
      z[i * 3 + 2] = fmaf(a0, R02, fmaf(a1, R12, a2 * R22));
    }

    // ---- drain my outstanding async stores (from n-1), then barrier so
    //      ALL waves' stores into this LDS tile are done before overwrite ----
#if USE_ASYNC
    wait_async0();
#endif
    __syncthreads();
#pragma unroll
    for (int k = 0; k < 9; ++k) lds[tid * 9 + k] = z[k];
    __syncthreads();
    {
      float4* __restrict__ gout4 =
          reinterpret_cast<float4*>(out + ((size_t)n * B + b0) * 9);
      const float4* lds4 = reinterpret_cast<const float4*>(lds);
#pragma unroll
      for (int i = 0; i < 3; ++i) {
        int idx = tid + i * TPB;
        if (idx < NV4) {
#if USE_ASYNC
          async_l2g_b128(&gout4[idx], &lds4[idx]);
#else
          gout4[idx] = lds4[idx];
#endif
        }
      }
    }
#if !USE_ASYNC
    __syncthreads();  // sync path: protect LDS reads before next overwrite
#endif
    // async path: next iteration's wait_async0 + barrier protects the tile,
    // and S_ENDPGM performs an implicit wait-idle for the final stores.
  }
}

// Fallback for B not divisible by TPB (not expected for the given shapes):
// one thread per (n, b), scalar access.
__global__ void so3_reparam_generic(const float* __restrict__ mu,
                                    const float* __restrict__ v,
                                    float* __restrict__ out, int B,
                                    long total) {
  long t = (long)blockIdx.x * blockDim.x + threadIdx.x;
  if (t >= total) return;
  int b = (int)(t % B);
  const float* mb = mu + (size_t)b * 9;
  float m[9];
#pragma unroll
  for (int k = 0; k < 9; ++k) m[k] = mb[k];

  const size_t vb = (size_t)t * 3;
  float vx = v[vb + 0], vy = v[vb + 1], vz = v[vb + 2];
  float d = fmaf(vx, vx, fmaf(vy, vy, vz * vz));
  float inv = __frsqrt_rn(d);
  float theta = d * inv;
  float ux = vx * inv, uy = vy * inv, uz = vz * inv;
  float s = __sinf(theta), c = __cosf(theta), oc = 1.0f - c;
  float ocx = oc * ux, ocy = oc * uy, ocz = oc * uz;
  float R00 = fmaf(ocx, ux, c);
  float R01 = fmaf(ocx, uy, -s * uz);
  float R02 = fmaf(ocx, uz, s * uy);
  float R10 = fmaf(ocy, ux, s * uz);
  float R11 = fmaf(ocy, uy, c);
  float R12 = fmaf(ocy, uz, -s * ux);
  float R20 = fmaf(ocz, ux, -s * uy);
  float R21 = fmaf(ocz, uy, s * ux);
  float R22 = fmaf(ocz, uz, c);

  float* zo = out + (size_t)t * 9;
#pragma unroll
  for (int i = 0; i < 3; ++i) {
    float a0 = m[i * 3 + 0], a1 = m[i * 3 + 1], a2 = m[i * 3 + 2];
    zo[i * 3 + 0] = fmaf(a0, R00, fmaf(a1, R10, a2 * R20));
    zo[i * 3 + 1] = fmaf(a0, R01, fmaf(a1, R11, a2 * R21));
    zo[i * 3 + 2] = fmaf(a0, R02, fmaf(a1, R12, a2 * R22));
  }
}

extern "C" void kernel_launch(void* const* d_in, const int* in_sizes, int n_in,
                              void* d_out, int out_size, void* d_ws,
                              size_t ws_size, hipStream_t stream) {
  (void)n_in;
  (void)d_ws;
  (void)ws_size;
  const float* mu = (const float*)d_in[0];  // [B, 3, 3]
  const float* v = (const float*)d_in[1];   // [N, B, 3]
  float* out = (float*)d_out;               // [N, B, 3, 3]

  const int B = in_sizes[0] / 9;
  const int nsamp = (int)((long)in_sizes[1] / ((long)B * 3));
  (void)out_size;

  if (B % TPB == 0) {
    so3_reparam_tiled<<<B / TPB, TPB, 0, stream>>>(mu, v, out, B, nsamp);
  } else {
    long total = (long)nsamp * B;
    int blocks = (int)((total + TPB - 1) / TPB);
    so3_reparam_generic<<<blocks, TPB, 0, stream>>>(mu, v, out, B, total);
  }
}